// WaveletKANLayer_81106162418067
// MI455X (gfx1250) — compile-verified
//
#include <hip/hip_runtime.h>

typedef float v2f __attribute__((ext_vector_type(2)));
typedef float v8f __attribute__((ext_vector_type(8)));

#define B_SZ   1024
#define IN_SZ  512
#define OUT_SZ 512

#define TM  64   // batch-tile rows per workgroup
#define TN  64   // out-tile cols per workgroup
#define KC  64   // IN staging chunk for wavelet phase
#define LDP 68   // padded LDS row stride (floats) to break bank conflicts

#define LOG2E    1.4426950408889634f
#define NHALF_L2 0.7213475204444817f   /* 0.5*log2(e) */
#define MH_CONST 0.8673250705840776f   /* 2/(sqrt(3)*pi^0.25) */

__launch_bounds__(256)
__global__ void waveletkan_kernel(const float* __restrict__ x,
                                  const float* __restrict__ scl,
                                  const float* __restrict__ trs,
                                  const float* __restrict__ bw,
                                  const float* __restrict__ ww,
                                  float* __restrict__ out) {
  __shared__ float base_t[TM][LDP];  // swish(x@W) tile
  __shared__ float x_t[TM][LDP];     // staged x chunk

  const int tid  = threadIdx.x;
  const int lane = tid & 31;
  const int wv   = tid >> 5;                  // wave id 0..7
  const int n0   = blockIdx.x * TN;           // out-col base
  const int b0   = blockIdx.y * TM;           // batch-row base

  // ============ Phase 1: base GEMM via V_WMMA_F32_16X16X4_F32 ============
  {
    const int wm    = wv & 3;                 // row subtile 0..3 (16 rows each)
    const int wnb   = (wv >> 2) * 2;          // col subtile base (0 or 2)
    const int m     = lane & 15;              // M (and N for B operand)
    const int klane = (lane >> 4) * 2;        // K pair base per ISA layout

    const float* aptr  = x  + (size_t)(b0 + wm * 16 + m) * IN_SZ + klane;
    const float* bptr0 = bw + (size_t)klane * OUT_SZ + (n0 + wnb * 16 + m);
    const float* bptr1 = bptr0 + 16;

    v8f c0 = {0.f, 0.f, 0.f, 0.f, 0.f, 0.f, 0.f, 0.f};
    v8f c1 = c0;

    for (int k = 0; k < IN_SZ; k += 4) {
      v2f a = *(const v2f*)(aptr + k);        // A[m][k+klane], A[m][k+klane+1]
      const int off = k * OUT_SZ;
      v2f bA, bB;
      bA.x = bptr0[off];
      bA.y = bptr0[off + OUT_SZ];
      bB.x = bptr1[off];
      bB.y = bptr1[off + OUT_SZ];
      c0 = __builtin_amdgcn_wmma_f32_16x16x4_f32(false, a, false, bA,
                                                 (short)0, c0, false, false);
      c1 = __builtin_amdgcn_wmma_f32_16x16x4_f32(false, a, false, bB,
                                                 (short)0, c1, false, false);
    }

    // swish(v) = v * sigmoid(v) ; sigmoid via v_exp_f32 + v_rcp_f32
    const int mrow = wm * 16 + 8 * (lane >> 4);
    const int ncol = lane & 15;
#pragma unroll
    for (int r = 0; r < 8; ++r) {
      float g0 = c0[r];
      float g1 = c1[r];
      float s0 = __builtin_amdgcn_rcpf(1.0f + __builtin_amdgcn_exp2f(g0 * -LOG2E));
      float s1 = __builtin_amdgcn_rcpf(1.0f + __builtin_amdgcn_exp2f(g1 * -LOG2E));
      base_t[mrow + r][ wnb      * 16 + ncol] = g0 * s0;
      base_t[mrow + r][(wnb + 1) * 16 + ncol] = g1 * s1;
    }
  }

  // ============ Phase 2: wavelet path (VALU/transcendental bound) ============
  const int o_l = tid & 63;                   // out column within tile
  const int bg  = tid >> 6;                   // batch group 0..3 (16 rows each)
  const int o   = n0 + o_l;

  const float* trow = trs + (size_t)o * IN_SZ;
  const float* srow = scl + (size_t)o * IN_SZ;
  const float* wrow = ww  + (size_t)o * IN_SZ;

  float acc[16];
#pragma unroll
  for (int r = 0; r < 16; ++r) acc[r] = 0.0f;

  for (int kc = 0; kc < IN_SZ; kc += KC) {
    __syncthreads();   // protect x_t from previous chunk's readers
#pragma unroll
    for (int p = 0; p < TM / 4; ++p) {
      const int row = bg + 4 * p;             // covers 0..63 across bg values
      x_t[row][o_l] = x[(size_t)(b0 + row) * IN_SZ + kc + o_l];
    }
    __syncthreads();

    for (int ii = 0; ii < KC; ++ii) {
      const float t  = trow[kc + ii];
      const float rs = __builtin_amdgcn_rcpf(srow[kc + ii]);
      const float wg = wrow[kc + ii];
#pragma unroll
      for (int r = 0; r < 16; ++r) {
        const float xv = x_t[bg * 16 + r][ii];     // same-addr broadcast per wave
        const float v  = (xv - t) * rs;
        const float sq = v * v;
        const float e  = __builtin_amdgcn_exp2f(sq * -NHALF_L2);
        acc[r] += wg * (sq - 1.0f) * e;
      }
    }
  }

  __syncthreads();

  // ============ Combine & store ============
#pragma unroll
  for (int r = 0; r < 16; ++r) {
    const int b_l = bg * 16 + r;
    out[(size_t)(b0 + b_l) * OUT_SZ + o] = base_t[b_l][o_l] + MH_CONST * acc[r];
  }
}

extern "C" void kernel_launch(void* const* d_in, const int* in_sizes, int n_in,
                              void* d_out, int out_size, void* d_ws, size_t ws_size,
                              hipStream_t stream) {
  (void)in_sizes; (void)n_in; (void)out_size; (void)d_ws; (void)ws_size;
  const float* x   = (const float*)d_in[0];
  const float* scl = (const float*)d_in[1];
  const float* trs = (const float*)d_in[2];
  const float* bw  = (const float*)d_in[3];
  const float* ww  = (const float*)d_in[4];
  float* out = (float*)d_out;

  dim3 grid(OUT_SZ / TN, B_SZ / TM);  // 8 x 16 workgroups
  hipLaunchKernelGGL(waveletkan_kernel, grid, dim3(256), 0, stream,
                     x, scl, trs, bw, ww, out);
}